// AdaptiveEmbedding_76038101008866
// MI455X (gfx1250) — compile-verified
//
#include <hip/hip_runtime.h>
#include <hip/hip_bf16.h>
#include <math.h>

#define B_SZ 256
#define R_SZ 36
#define T_SZ 64
#define D_SZ 1024
#define EPS_L2 1e-8f
#define EPS_BN 1e-5f
#define LEAK 0.1f
#define KC 64

typedef __attribute__((ext_vector_type(2))) float v2f;
typedef __attribute__((ext_vector_type(8))) float v8f;

// ---------------------------------------------------------------- utilities
__global__ void zero_ws_kernel(float* p, int n) {
    int i = blockIdx.x * blockDim.x + threadIdx.x;
    if (i < n) p[i] = 0.f;
}

// ---------------------------------------------------------------- cap path
// cap_embed (B,T,D) -> cap_mean (B,D):
// y = lrelu(x); n = sqrt(sum_t y^2)+eps; cap_mean = (sum_{t<len} y) / (n*len)
__global__ void cap_pre_kernel(const float* __restrict__ cap,
                               const int* __restrict__ lens,
                               float* __restrict__ cap_mean) {
    const int b = blockIdx.x;
    const int len = lens[b];
    const float* cb = cap + (size_t)b * T_SZ * D_SZ;
    float sumsq[4] = {0.f, 0.f, 0.f, 0.f};
    float summ[4]  = {0.f, 0.f, 0.f, 0.f};
    for (int t = 0; t < T_SZ; ++t) {
        const float* row = cb + (size_t)t * D_SZ;
        const bool valid = (t < len);
#pragma unroll
        for (int i = 0; i < 4; ++i) {
            float x = row[threadIdx.x + i * 256];
            float y = x > 0.f ? x : LEAK * x;
            sumsq[i] += y * y;
            if (valid) summ[i] += y;
        }
    }
#pragma unroll
    for (int i = 0; i < 4; ++i) {
        int d = threadIdx.x + i * 256;
        float n = sqrtf(sumsq[i]) + EPS_L2;
        cap_mean[(size_t)b * D_SZ + d] = summ[i] / (n * (float)len);
    }
}

// ---------------------------------------------------------------- img path
// img_embed (B,R,D) -> meanR (B,D), plus global per-channel sums of the
// normalized values (colsum1) and their squares (colsum2) for BN stats.
__global__ void img_pre_kernel(const float* __restrict__ img,
                               float* __restrict__ meanR,
                               float* __restrict__ cs1,
                               float* __restrict__ cs2) {
    const int b = blockIdx.x;
    const float* ib = img + (size_t)b * R_SZ * D_SZ;
    float sumsq[4] = {0.f, 0.f, 0.f, 0.f};
    float summ[4]  = {0.f, 0.f, 0.f, 0.f};
    for (int r = 0; r < R_SZ; ++r) {
        const float* row = ib + (size_t)r * D_SZ;
#pragma unroll
        for (int i = 0; i < 4; ++i) {
            float x = row[threadIdx.x + i * 256];
            float y = x > 0.f ? x : LEAK * x;
            sumsq[i] += y * y;
            summ[i]  += y;
        }
    }
#pragma unroll
    for (int i = 0; i < 4; ++i) {
        int d = threadIdx.x + i * 256;
        float n = sqrtf(sumsq[i]) + EPS_L2;
        float s1 = summ[i] / n;          // sum_r img_n
        float s2 = sumsq[i] / (n * n);   // sum_r img_n^2
        meanR[(size_t)b * D_SZ + d] = s1 * (1.f / (float)R_SZ);
        atomicAdd(cs1 + d, s1);
        atomicAdd(cs2 + d, s2);
    }
}

// base[b,d] = (meanR - mu)*gamma/sqrt(var+eps) + beta ; base2 = base^2
__global__ void make_base_kernel(const float* __restrict__ meanR,
                                 const float* __restrict__ cs1,
                                 const float* __restrict__ cs2,
                                 const float* __restrict__ bn_w,
                                 const float* __restrict__ bn_b,
                                 float* __restrict__ base,
                                 float* __restrict__ base2) {
    const int b = blockIdx.x;
    const float inv_n = 1.f / (float)(B_SZ * R_SZ);
#pragma unroll
    for (int i = 0; i < 4; ++i) {
        int d = threadIdx.x + i * 256;
        float mu  = cs1[d] * inv_n;
        float ex2 = cs2[d] * inv_n;
        float var = ex2 - mu * mu;
        float scale = bn_w[d] * rsqrtf(var + EPS_BN);
        float v = (meanR[(size_t)b * D_SZ + d] - mu) * scale + bn_b[d];
        base[(size_t)b * D_SZ + d]  = v;
        base2[(size_t)b * D_SZ + d] = v * v;
    }
}

// ---------------------------------------------------------------- fc GEMM
// params (256 x 2048) = cap_mean (256x1024) @ fc_w^T (1024x2048) + fc_b
// C[m,n] = sum_k A[m,k]*Bw[n,k]  -> row-dot-row, WMMA f32 16x16x4.
__global__ __launch_bounds__(128) void fc_gemm_kernel(
        const float* __restrict__ A, const float* __restrict__ Bw,
        const float* __restrict__ bias, float* __restrict__ Cout) {
    __shared__ float sA[2][16][KC + 1];
    __shared__ float sB[2][16][KC + 1];
    const int tid  = threadIdx.x;
    const int wave = tid >> 5;
    const int lane = tid & 31;
    const int wm = wave >> 1, wn = wave & 1;
    const int half = lane >> 4, fr = lane & 15;
    const int m0 = blockIdx.y * 32;
    const int n0 = blockIdx.x * 32;
    v8f acc = {};
    for (int k0 = 0; k0 < D_SZ; k0 += KC) {
        for (int idx = tid; idx < 2 * 16 * KC; idx += 128) {
            int blk = idx / (16 * KC);
            int rem = idx % (16 * KC);
            int row = rem / KC, col = rem % KC;
            sA[blk][row][col] = A [(size_t)(m0 + blk * 16 + row) * D_SZ + k0 + col];
            sB[blk][row][col] = Bw[(size_t)(n0 + blk * 16 + row) * D_SZ + k0 + col];
        }
        __syncthreads();
#pragma unroll
        for (int kk = 0; kk < KC; kk += 4) {
            int kb = kk + half * 2;
            v2f a, b;
            a[0] = sA[wm][fr][kb]; a[1] = sA[wm][fr][kb + 1];
            b[0] = sB[wn][fr][kb]; b[1] = sB[wn][fr][kb + 1];
            acc = __builtin_amdgcn_wmma_f32_16x16x4_f32(
                    false, a, false, b, (short)0, acc, false, false);
        }
        __syncthreads();
    }
#pragma unroll
    for (int j = 0; j < 8; ++j) {
        int m = m0 + wm * 16 + j + half * 8;
        int n = n0 + wn * 16 + fr;
        Cout[(size_t)m * (2 * D_SZ) + n] = acc[j] + bias[n];
    }
}

// ---------------------------------------------------------------- P/A2/AB + q,r
// cap_hat = cap_mean/(||row||+eps); alpha=params[c,2d], beta=params[c,2d+1]
__global__ void make_pab_kernel(const float* __restrict__ cap_mean,
                                const float* __restrict__ params,
                                float* __restrict__ P, float* __restrict__ A2m,
                                float* __restrict__ ABm,
                                float* __restrict__ qv, float* __restrict__ rv) {
    const int c = blockIdx.x;
    const int tid = threadIdx.x;
    __shared__ float red[256];

    float ss = 0.f;
#pragma unroll
    for (int i = 0; i < 4; ++i) {
        float x = cap_mean[(size_t)c * D_SZ + tid + i * 256];
        ss += x * x;
    }
    red[tid] = ss; __syncthreads();
    for (int s = 128; s > 0; s >>= 1) {
        if (tid < s) red[tid] += red[tid + s];
        __syncthreads();
    }
    const float nrm = sqrtf(red[0]) + EPS_L2;
    __syncthreads();

    float q = 0.f, r = 0.f;
#pragma unroll
    for (int i = 0; i < 4; ++i) {
        int d = tid + i * 256;
        float alpha = params[(size_t)c * (2 * D_SZ) + 2 * d];
        float beta  = params[(size_t)c * (2 * D_SZ) + 2 * d + 1];
        float ch = cap_mean[(size_t)c * D_SZ + d] / nrm;
        P  [(size_t)c * D_SZ + d] = alpha * ch;
        A2m[(size_t)c * D_SZ + d] = alpha * alpha;
        ABm[(size_t)c * D_SZ + d] = alpha * beta;
        q += beta * ch;
        r += beta * beta;
    }
    red[tid] = q; __syncthreads();
    for (int s = 128; s > 0; s >>= 1) {
        if (tid < s) red[tid] += red[tid + s];
        __syncthreads();
    }
    if (tid == 0) qv[c] = red[0];
    __syncthreads();
    red[tid] = r; __syncthreads();
    for (int s = 128; s > 0; s >>= 1) {
        if (tid < s) red[tid] += red[tid + s];
        __syncthreads();
    }
    if (tid == 0) rv[c] = red[0];
}

// ---------------------------------------------------------------- sims GEMM
// Three fused 256x256x1024 GEMMs + normalization epilogue:
// sims[b,c] = (<base[b],P[c]> + q[c]) /
//             (sqrt(<base2[b],A2[c]> + 2<base[b],AB[c]> + r[c]) + eps)
__global__ __launch_bounds__(128) void sims_gemm_kernel(
        const float* __restrict__ base, const float* __restrict__ base2,
        const float* __restrict__ P, const float* __restrict__ A2m,
        const float* __restrict__ ABm,
        const float* __restrict__ qv, const float* __restrict__ rv,
        float* __restrict__ out) {
    __shared__ float sBa [2][16][KC + 1];
    __shared__ float sBa2[2][16][KC + 1];
    __shared__ float sP  [2][16][KC + 1];
    __shared__ float sA2 [2][16][KC + 1];
    __shared__ float sAB [2][16][KC + 1];
    const int tid  = threadIdx.x;
    const int wave = tid >> 5;
    const int lane = tid & 31;
    const int wm = wave >> 1, wn = wave & 1;
    const int half = lane >> 4, fr = lane & 15;
    const int bm0 = blockIdx.y * 32;   // img rows
    const int cn0 = blockIdx.x * 32;   // cap rows
    v8f acc1 = {}, acc2 = {}, acc3 = {};
    for (int k0 = 0; k0 < D_SZ; k0 += KC) {
        for (int idx = tid; idx < 2 * 16 * KC; idx += 128) {
            int blk = idx / (16 * KC);
            int rem = idx % (16 * KC);
            int row = rem / KC, col = rem % KC;
            size_t ga = (size_t)(bm0 + blk * 16 + row) * D_SZ + k0 + col;
            size_t gb = (size_t)(cn0 + blk * 16 + row) * D_SZ + k0 + col;
            sBa [blk][row][col] = base [ga];
            sBa2[blk][row][col] = base2[ga];
            sP  [blk][row][col] = P  [gb];
            sA2 [blk][row][col] = A2m[gb];
            sAB [blk][row][col] = ABm[gb];
        }
        __syncthreads();
#pragma unroll
        for (int kk = 0; kk < KC; kk += 4) {
            int kb = kk + half * 2;
            v2f a1, a2, bp, b2, bb;
            a1[0] = sBa [wm][fr][kb]; a1[1] = sBa [wm][fr][kb + 1];
            a2[0] = sBa2[wm][fr][kb]; a2[1] = sBa2[wm][fr][kb + 1];
            bp[0] = sP  [wn][fr][kb]; bp[1] = sP  [wn][fr][kb + 1];
            b2[0] = sA2 [wn][fr][kb]; b2[1] = sA2 [wn][fr][kb + 1];
            bb[0] = sAB [wn][fr][kb]; bb[1] = sAB [wn][fr][kb + 1];
            acc1 = __builtin_amdgcn_wmma_f32_16x16x4_f32(
                    false, a1, false, bp, (short)0, acc1, false, false);
            acc2 = __builtin_amdgcn_wmma_f32_16x16x4_f32(
                    false, a2, false, b2, (short)0, acc2, false, false);
            acc3 = __builtin_amdgcn_wmma_f32_16x16x4_f32(
                    false, a1, false, bb, (short)0, acc3, false, false);
        }
        __syncthreads();
    }
#pragma unroll
    for (int j = 0; j < 8; ++j) {
        int b = bm0 + wm * 16 + j + half * 8;
        int c = cn0 + wn * 16 + fr;
        float num = acc1[j] + qv[c];
        float den = sqrtf(acc2[j] + 2.f * acc3[j] + rv[c]) + EPS_L2;
        out[(size_t)b * B_SZ + c] = num / den;
    }
}

// ---------------------------------------------------------------- launch
extern "C" void kernel_launch(void* const* d_in, const int* in_sizes, int n_in,
                              void* d_out, int out_size, void* d_ws, size_t ws_size,
                              hipStream_t stream) {
    const float* img   = (const float*)d_in[0];   // (B,R,D)
    const float* cap   = (const float*)d_in[1];   // (B,T,D)
    const int*   lens  = (const int*)  d_in[2];   // (B,)
    const float* fc_w  = (const float*)d_in[3];   // (2D,D)
    const float* fc_b  = (const float*)d_in[4];   // (2D,)
    const float* bn_w  = (const float*)d_in[5];   // (D,)
    const float* bn_b  = (const float*)d_in[6];   // (D,)
    float* out = (float*)d_out;                   // (B_img, B_cap) = (256,256)

    float* ws = (float*)d_ws;
    const size_t BD = (size_t)B_SZ * D_SZ;        // 262144
    float* cap_mean = ws;                 // BD
    float* meanR    = cap_mean + BD;      // BD
    float* base     = meanR + BD;         // BD
    float* base2    = base + BD;          // BD
    float* P        = base2 + BD;         // BD
    float* A2m      = P + BD;             // BD
    float* ABm      = A2m + BD;           // BD
    float* params   = ABm + BD;           // 2*BD
    float* colsum1  = params + 2 * BD;    // D
    float* colsum2  = colsum1 + D_SZ;     // D
    float* qv       = colsum2 + D_SZ;     // B
    float* rv       = qv + B_SZ;          // B

    zero_ws_kernel<<<dim3((2 * D_SZ + 255) / 256), 256, 0, stream>>>(colsum1, 2 * D_SZ);
    cap_pre_kernel<<<dim3(B_SZ), 256, 0, stream>>>(cap, lens, cap_mean);
    img_pre_kernel<<<dim3(B_SZ), 256, 0, stream>>>(img, meanR, colsum1, colsum2);
    make_base_kernel<<<dim3(B_SZ), 256, 0, stream>>>(meanR, colsum1, colsum2,
                                                     bn_w, bn_b, base, base2);
    fc_gemm_kernel<<<dim3(2 * D_SZ / 32, B_SZ / 32), 128, 0, stream>>>(
            cap_mean, fc_w, fc_b, params);
    make_pab_kernel<<<dim3(B_SZ), 256, 0, stream>>>(cap_mean, params,
                                                    P, A2m, ABm, qv, rv);
    sims_gemm_kernel<<<dim3(B_SZ / 32, B_SZ / 32), 128, 0, stream>>>(
            base, base2, P, A2m, ABm, qv, rv, out);
}